// MambaBlock_33354716021470
// MI455X (gfx1250) — compile-verified
//
#include <hip/hip_runtime.h>
#include <hip/hip_bf16.h>
#include <math.h>

// ---------------- problem constants ----------------
#define D_MODEL 1024
#define D_INNER 2048
#define D_STATE 16
#define D_CONV  4
#define DT_RANK 64
#define BATCH   2
#define SEQLEN  1024
#define NTOK    (BATCH * SEQLEN)          // 2048 rows for all GEMMs
#define EPSF    1e-5f

// workspace layout (float elements)
#define OFF_HS   ((size_t)0)                                   // 2048*1024
#define OFF_XZ   (OFF_HS   + (size_t)NTOK * D_MODEL)           // 2048*4096
#define OFF_XC   (OFF_XZ   + (size_t)NTOK * 2 * D_INNER)       // 2048*2048
#define OFF_XDBL (OFF_XC   + (size_t)NTOK * D_INNER)           // 2048*96
#define OFF_DT   (OFF_XDBL + (size_t)NTOK * (DT_RANK + 2*D_STATE))
#define OFF_Y    (OFF_DT   + (size_t)NTOK * D_INNER)

typedef __attribute__((ext_vector_type(2))) float v2f;
typedef __attribute__((ext_vector_type(8))) float v8f;

__device__ __forceinline__ float silu_f(float x) {
    return x / (1.0f + __expf(-x));
}

// ---------------------------------------------------------------------------
// Kernel 1: res = hs + residual ; hs_norm = LayerNorm(res)*w + b
// ---------------------------------------------------------------------------
__global__ void add_layernorm_kernel(const float* __restrict__ hs_in,
                                     const float* __restrict__ res_in,
                                     const float* __restrict__ w,
                                     const float* __restrict__ bias,
                                     float* __restrict__ res_out,
                                     float* __restrict__ hs_out) {
    const int row = blockIdx.x;          // 0 .. NTOK-1
    const int tid = threadIdx.x;         // 0 .. 255
    const size_t base = (size_t)row * D_MODEL;

    float v[4];
    float s = 0.0f, sq = 0.0f;
#pragma unroll
    for (int i = 0; i < 4; ++i) {
        const int j = tid + 256 * i;
        const float x = hs_in[base + j] + res_in[base + j];
        v[i] = x;
        s += x;
        sq += x * x;
        res_out[base + j] = x;
    }

    __shared__ float rs[256];
    __shared__ float rq[256];
    rs[tid] = s;
    rq[tid] = sq;
    __syncthreads();
    for (int off = 128; off > 0; off >>= 1) {
        if (tid < off) {
            rs[tid] += rs[tid + off];
            rq[tid] += rq[tid + off];
        }
        __syncthreads();
    }
    const float mu   = rs[0] * (1.0f / (float)D_MODEL);
    const float var  = rq[0] * (1.0f / (float)D_MODEL) - mu * mu;
    const float rstd = rsqrtf(var + EPSF);

#pragma unroll
    for (int i = 0; i < 4; ++i) {
        const int j = tid + 256 * i;
        hs_out[base + j] = (v[i] - mu) * rstd * w[j] + bias[j];
    }
}

// ---------------------------------------------------------------------------
// Kernel 2: fp32 WMMA GEMM, branchless tiling + explicit 2-stage software
// pipeline (double-buffered fragments) so next stage's global loads are in
// flight while the current stage's V_WMMA_F32_16X16X4_F32 issue -> partial
// s_wait_loadcnt instead of full drains.
//   grid.y * 8 waves : M tiles of 16 rows    (M = grid.y * 128 exactly)
//   grid.x           : N supertiles of TN*16 (N = grid.x * TN*16 exactly)
// steps = K/4 must be even (K in {64, 1024, 2048} -> 16/256/512). 
// MODE 0: plain store   MODE 1: val = softplus(val + bias[col])
// ---------------------------------------------------------------------------
template <int TN, int MODE>
__global__ void gemm_wmma_f32(const float* __restrict__ A, int lda,
                              const float* __restrict__ Bw, int ldb,
                              float* __restrict__ C, int ldc,
                              int K, const float* __restrict__ bias) {
    const int lane = threadIdx.x & 31;
    const int wv   = threadIdx.x >> 5;                 // 0..7
    const int m0   = (blockIdx.y * 8 + wv) << 4;
    const int n0   = blockIdx.x * (TN * 16);

    // A fragment (16x4 f32): lanes 0-15 -> M=lane, K=+0,+1 ; lanes 16-31 -> K=+2,+3
    const int arow = m0 + (lane & 15);
    const int koff = (lane >> 4) << 1;
    const int bcol = lane & 15;

    const float* ap = A  + (size_t)arow * lda + koff;
    const float* bp = Bw + (size_t)koff * ldb + n0 + bcol;

    v8f acc[TN] = {};

    v2f a0, a1;
    v2f b0[TN], b1[TN];

#define LOAD_A(dst, p)                                                         \
    do { (dst).x = (p)[0]; (dst).y = (p)[1]; } while (0)
#define LOAD_B(dst, p)                                                         \
    do {                                                                       \
        _Pragma("unroll") for (int t = 0; t < TN; ++t) {                       \
            (dst)[t].x = (p)[t * 16];                                          \
            (dst)[t].y = (p)[t * 16 + (size_t)ldb];                            \
        }                                                                      \
    } while (0)
#define WMMA_STAGE(af, bf)                                                     \
    do {                                                                       \
        _Pragma("unroll") for (int t = 0; t < TN; ++t) {                       \
            acc[t] = __builtin_amdgcn_wmma_f32_16x16x4_f32(                    \
                false, (af), false, (bf)[t], (short)0, acc[t], false, false);  \
        }                                                                      \
    } while (0)

    const int steps = K >> 2;                 // even by construction

    // prologue: stage 0 -> buf0
    LOAD_A(a0, ap);
    LOAD_B(b0, bp);

    // main pipeline: per iteration handle stages s and s+1
    for (int s = 0; s + 2 < steps; s += 2) {
        const float* ap1 = ap + (size_t)(s + 1) * 4;
        const float* bp1 = bp + (size_t)(s + 1) * 4 * ldb;
        LOAD_A(a1, ap1);                       // loads for stage s+1 in flight
        LOAD_B(b1, bp1);
        __builtin_prefetch(ap1 + 32, 0, 0);    // 8 k-steps ahead
        __builtin_prefetch(bp1 + (size_t)32 * ldb, 0, 0);
        WMMA_STAGE(a0, b0);                    // compute stage s

        const float* ap2 = ap + (size_t)(s + 2) * 4;
        const float* bp2 = bp + (size_t)(s + 2) * 4 * ldb;
        LOAD_A(a0, ap2);                       // loads for stage s+2 in flight
        LOAD_B(b0, bp2);
        WMMA_STAGE(a1, b1);                    // compute stage s+1
    }

    // epilogue: stages steps-2 (already in buf0) and steps-1
    {
        const float* ap1 = ap + (size_t)(steps - 1) * 4;
        const float* bp1 = bp + (size_t)(steps - 1) * 4 * ldb;
        LOAD_A(a1, ap1);
        LOAD_B(b1, bp1);
        WMMA_STAGE(a0, b0);
        WMMA_STAGE(a1, b1);
    }

#undef LOAD_A
#undef LOAD_B
#undef WMMA_STAGE

    // C/D fragment: VGPR i -> row rbase+i
    const int rbase = m0 + ((lane >> 4) << 3);
#pragma unroll
    for (int t = 0; t < TN; ++t) {
        const int col = n0 + t * 16 + bcol;
        float* cp = C + (size_t)rbase * ldc + col;
#pragma unroll
        for (int i = 0; i < 8; ++i) {
            float val = acc[t][i];
            if (MODE == 1) {
                val += bias[col];
                val = (val > 0.0f) ? (val + log1pf(__expf(-val)))
                                   : log1pf(__expf(val));
            }
            cp[(size_t)i * ldc] = val;
        }
    }
}

// ---------------------------------------------------------------------------
// Kernel 3: causal depthwise conv (k=4) + bias + SiLU
// ---------------------------------------------------------------------------
__global__ void conv_silu_kernel(const float* __restrict__ xz,
                                 const float* __restrict__ cw,
                                 const float* __restrict__ cb,
                                 float* __restrict__ xc) {
    const int idx = blockIdx.x * blockDim.x + threadIdx.x;  // B*L*D_INNER
    if (idx >= NTOK * D_INNER) return;
    const int c  = idx & (D_INNER - 1);
    const int bl = idx >> 11;                 // b*L + l
    const int l  = bl & (SEQLEN - 1);

    float acc = cb[c];
#pragma unroll
    for (int k = 0; k < D_CONV; ++k) {
        const int ls = l + k - (D_CONV - 1);
        if (ls >= 0) {
            acc += xz[(size_t)(bl + k - (D_CONV - 1)) * (2 * D_INNER) + c]
                   * cw[c * D_CONV + k];
        }
    }
    xc[idx] = silu_f(acc);
}

// ---------------------------------------------------------------------------
// Kernel 4: selective scan, one thread per (batch, channel), 16-state in regs,
// B/C rows broadcast through LDS. Fused: y = (scan_y + xc*D) * silu(z)
// ---------------------------------------------------------------------------
__global__ void scan_kernel(const float* __restrict__ xdbl,
                            const float* __restrict__ dt,
                            const float* __restrict__ xc,
                            const float* __restrict__ xz,
                            const float* __restrict__ A_log,
                            const float* __restrict__ Dvec,
                            float* __restrict__ y) {
    const int b = blockIdx.x >> 3;
    const int c = ((blockIdx.x & 7) << 8) + threadIdx.x;     // 0..D_INNER-1

    float a[D_STATE];
#pragma unroll
    for (int n = 0; n < D_STATE; ++n)
        a[n] = -__expf(A_log[c * D_STATE + n]);
    const float Dc = Dvec[c];

    float h[D_STATE] = {};
    __shared__ float sB[D_STATE];
    __shared__ float sC[D_STATE];

    for (int l = 0; l < SEQLEN; ++l) {
        const size_t bl = (size_t)b * SEQLEN + l;
        if (threadIdx.x < 2 * D_STATE) {
            const int n  = threadIdx.x;
            const float v = xdbl[bl * (DT_RANK + 2 * D_STATE) + DT_RANK + n];
            if (n < D_STATE) sB[n] = v;
            else             sC[n - D_STATE] = v;
        }
        __syncthreads();

        const float dtv = dt[bl * D_INNER + c];
        const float xv  = xc[bl * D_INNER + c];
        float yv = 0.0f;
#pragma unroll
        for (int n = 0; n < D_STATE; ++n) {
            const float dA = __expf(dtv * a[n]);
            h[n] = dA * h[n] + dtv * sB[n] * xv;
            yv += h[n] * sC[n];
        }
        yv += xv * Dc;
        const float zv = xz[bl * (2 * D_INNER) + D_INNER + c];
        yv *= silu_f(zv);
        y[bl * D_INNER + c] = yv;
        __syncthreads();
    }
}

// ---------------------------------------------------------------------------
// launcher
// ---------------------------------------------------------------------------
extern "C" void kernel_launch(void* const* d_in, const int* in_sizes, int n_in,
                              void* d_out, int out_size, void* d_ws, size_t ws_size,
                              hipStream_t stream) {
    const float* hidden    = (const float*)d_in[0];
    const float* residual  = (const float*)d_in[1];
    const float* norm_w    = (const float*)d_in[2];
    const float* norm_b    = (const float*)d_in[3];
    const float* in_proj_w = (const float*)d_in[4];   // (1024, 4096)
    const float* conv_w    = (const float*)d_in[5];   // (2048, 4)
    const float* conv_b    = (const float*)d_in[6];
    const float* x_proj_w  = (const float*)d_in[7];   // (2048, 96)
    const float* dt_proj_w = (const float*)d_in[8];   // (64, 2048)
    const float* dt_proj_b = (const float*)d_in[9];
    const float* A_log     = (const float*)d_in[10];  // (2048, 16)
    const float* Dvec      = (const float*)d_in[11];
    const float* out_proj_w= (const float*)d_in[12];  // (2048, 1024)

    float* ws   = (float*)d_ws;
    float* hs   = ws + OFF_HS;
    float* xz   = ws + OFF_XZ;
    float* xc   = ws + OFF_XC;
    float* xdbl = ws + OFF_XDBL;
    float* dtb  = ws + OFF_DT;
    float* ybuf = ws + OFF_Y;

    float* out_main = (float*)d_out;                       // (B,L,D_MODEL)
    float* res_out  = out_main + (size_t)NTOK * D_MODEL;   // second tuple element

    // 1) residual add + layernorm
    add_layernorm_kernel<<<NTOK, 256, 0, stream>>>(hidden, residual, norm_w,
                                                   norm_b, res_out, hs);

    // 2) xz = hs @ in_proj_w     (2048x1024)*(1024x4096), grid (64,16)
    gemm_wmma_f32<4, 0><<<dim3(64, 16), 256, 0, stream>>>(
        hs, D_MODEL, in_proj_w, 2 * D_INNER, xz, 2 * D_INNER,
        D_MODEL, nullptr);

    // 3) causal conv + silu on x half
    {
        const int n = NTOK * D_INNER;
        conv_silu_kernel<<<(n + 255) / 256, 256, 0, stream>>>(xz, conv_w,
                                                              conv_b, xc);
    }

    // 4) x_dbl = xc @ x_proj_w   (2048x2048)*(2048x96): TN=6 covers N=96
    gemm_wmma_f32<6, 0><<<dim3(1, 16), 256, 0, stream>>>(
        xc, D_INNER, x_proj_w, DT_RANK + 2 * D_STATE,
        xdbl, DT_RANK + 2 * D_STATE, D_INNER, nullptr);

    // 5) dt = softplus(x_dbl[:, :64] @ dt_proj_w + dt_proj_b), grid (32,16)
    gemm_wmma_f32<4, 1><<<dim3(32, 16), 256, 0, stream>>>(
        xdbl, DT_RANK + 2 * D_STATE, dt_proj_w, D_INNER,
        dtb, D_INNER, DT_RANK, dt_proj_b);

    // 6) selective scan + gating
    scan_kernel<<<BATCH * (D_INNER / 256), 256, 0, stream>>>(
        xdbl, dtb, xc, xz, A_log, Dvec, ybuf);

    // 7) out = y @ out_proj_w    (2048x2048)*(2048x1024), grid (16,16)
    gemm_wmma_f32<4, 0><<<dim3(16, 16), 256, 0, stream>>>(
        ybuf, D_INNER, out_proj_w, D_MODEL, out_main, D_MODEL,
        D_INNER, nullptr);
}